// HashEmbedding_37623913513633
// MI455X (gfx1250) — compile-verified
//
#include <hip/hip_runtime.h>
#include <cstdint>
#include <cstddef>

// HashEmbedding (EmbeddingBag mode='sum', per-sample weights) on MI455X/gfx1250.
//
// x:            [32, 2048, 3] int32 hashes in [0, 100000)
// emb_table:    [10001, 256]  f32  (10.25 MB -> L2 resident)
// weight_table: [300003, 1]   f32  ( 1.2 MB  -> L2 resident)
// out:          [32, 2048, 256] f32 (64 MB write-once -> non-temporal stores)
//
// One wave per token. The 3-row embedding gather per token is done by the
// Tensor Data Mover in GATHER mode (32-bit row indices, tile = 3 x 256 f32)
// into an LDS double buffer, software-pipelined one token ahead and synced
// with TENSORcnt. The wave itself only does scalar (SMEM) index/weight loads,
// ds_load_b128 reads from LDS, v_fma_f32, and non-temporal b128 stores.

typedef float        f4   __attribute__((ext_vector_type(4)));
typedef unsigned int u32x4 __attribute__((ext_vector_type(4)));
typedef int          i32x4 __attribute__((ext_vector_type(4)));
typedef int          i32x8 __attribute__((ext_vector_type(8)));

#define NUM_HASHES 3
#define EMBED_DIM  256
#define KVOCAB     100000
#define RATIO      10
#define NROWS      10001      // emb_table rows
#define WAVES_PER_BLK 8

__device__ __forceinline__ f4 fma4(f4 a, float w, f4 acc) {
    f4 r;
    r.x = fmaf(a.x, w, acc.x);
    r.y = fmaf(a.y, w, acc.y);
    r.z = fmaf(a.z, w, acc.z);
    r.w = fmaf(a.w, w, acc.w);
    return r;
}

__device__ __forceinline__ f4 mul4(f4 a, float w) {
    f4 r;
    r.x = a.x * w;
    r.y = a.y * w;
    r.z = a.z * w;
    r.w = a.w * w;
    return r;
}

// TDM gather of 3 embedding rows (256 f32 each) into LDS at lds_addr.
// D# per CDNA5 ISA 8.3/8.4/8.7:
//   group0: count=1, gather_mode=1, 32-bit indices, lds_addr, global_addr, type=2
//   group1: data_size=4B, tensor_dim0=256, tensor_dim1=10001,
//           tile_dim0=256, tile_dim1=#indices=3, tensor_dim0_stride=256
//   group2: row_index_0..2 (32-bit mode), group3: unused
__device__ __forceinline__ void tdm_gather3(const float* __restrict__ emb,
                                            uint32_t lds_addr,
                                            int r0, int r1, int r2) {
    const uint64_t ga = (uint64_t)(uintptr_t)emb;

    u32x4 g0;
    g0.x = 0xC0000001u;                               // count=1 | gather_mode | 32b idx
    g0.y = lds_addr;                                  // LDS byte address
    g0.z = (uint32_t)(ga & 0xFFFFFFFFu);              // global_addr[31:0]
    g0.w = (uint32_t)((ga >> 32) & 0x01FFFFFFu)       // global_addr[56:32]
         | (2u << 30);                                // type = 2 ("image")

    i32x8 g1;
    g1[0] = 0x00020000;                               // data_size = 2 -> 4 bytes
    g1[1] = (int)(256u   << 16);                      // tensor_dim0[15:0]  = 256
    g1[2] = (int)(10001u << 16);                      // tensor_dim1[15:0]  = 10001
    g1[3] = (int)(256u   << 16);                      // tile_dim0          = 256
    g1[4] = 3;                                        // tile_dim1 = #valid indices
    g1[5] = 256;                                      // tensor_dim0_stride = 256
    g1[6] = 0;
    g1[7] = 0;

    i32x4 g2; g2.x = r0; g2.y = r1; g2.z = r2; g2.w = 0;   // row indices
    i32x4 g3; g3.x = 0;  g3.y = 0;  g3.z = 0;  g3.w = 0;

#if defined(__clang_major__) && (__clang_major__ >= 23)
    i32x8 gpad; gpad[0]=0; gpad[1]=0; gpad[2]=0; gpad[3]=0;
    gpad[4]=0; gpad[5]=0; gpad[6]=0; gpad[7]=0;
    __builtin_amdgcn_tensor_load_to_lds(g0, g1, g2, g3, gpad, 0);
#else
    __builtin_amdgcn_tensor_load_to_lds(g0, g1, g2, g3, 0);
#endif
}

__global__ __launch_bounds__(256) void hash_emb_bag_kernel(
    const int*   __restrict__ x,       // [ntokens * 3]
    const float* __restrict__ emb,     // [10001 * 256]
    const float* __restrict__ wt,      // [300003]
    float*       __restrict__ out,     // [ntokens * 256]
    int ntokens)
{
    // Double-buffered gather staging: [buf][wave][3 rows * 256 f32] = 48 KB.
    __shared__ float lds[2][WAVES_PER_BLK][NUM_HASHES * EMBED_DIM];

    const int lane   = threadIdx.x & 31;
    const int wIdx   = threadIdx.x >> 5;
    const int nwaves = gridDim.x * WAVES_PER_BLK;

    // Force the token id into an SGPR: index/weight loads become SMEM,
    // and the TDM descriptor operands are wave-uniform SGPR groups.
    int t = __builtin_amdgcn_readfirstlane((int)(blockIdx.x * WAVES_PER_BLK + wIdx));
    if (t >= ntokens) return;

    // LDS byte addresses of this wave's two staging buffers (low 32 bits of
    // the shared-aperture generic address == LDS byte offset).
    const uint32_t ldsA = __builtin_amdgcn_readfirstlane(
        (int)(uint32_t)(uintptr_t)&lds[0][wIdx][0]);
    const uint32_t ldsB = __builtin_amdgcn_readfirstlane(
        (int)(uint32_t)(uintptr_t)&lds[1][wIdx][0]);

    // First token's hash ids (scalar s_load_b96) + kick off its TDM gather.
    int i0 = __builtin_amdgcn_readfirstlane(x[t * 3 + 0]);
    int i1 = __builtin_amdgcn_readfirstlane(x[t * 3 + 1]);
    int i2 = __builtin_amdgcn_readfirstlane(x[t * 3 + 2]);
    tdm_gather3(emb, ldsA, i0 / RATIO, i1 / RATIO, i2 / RATIO);

    int cur = 0;
    while (true) {
        const int  tn   = t + nwaves;
        const bool more = (tn < ntokens);

        // Pipeline: fetch next token's ids and launch its gather into the
        // other buffer before consuming the current one.
        int n0 = 0, n1 = 0, n2 = 0;
        if (more) {
            n0 = __builtin_amdgcn_readfirstlane(x[tn * 3 + 0]);
            n1 = __builtin_amdgcn_readfirstlane(x[tn * 3 + 1]);
            n2 = __builtin_amdgcn_readfirstlane(x[tn * 3 + 2]);
            tdm_gather3(emb, cur ? ldsA : ldsB,
                        n0 / RATIO, n1 / RATIO, n2 / RATIO);
        }

        // Per-sample weights (scalar SMEM loads), salted per hash slot.
        const float w0 = wt[i0];
        const float w1 = wt[i1 + (KVOCAB + 1)];
        const float w2 = wt[i2 + 2 * (KVOCAB + 1)];

        // TDM completes in-order per wave: tensorcnt<=1 means the gather for
        // the CURRENT buffer has landed in LDS.
        if (more) __builtin_amdgcn_s_wait_tensorcnt(1);
        else      __builtin_amdgcn_s_wait_tensorcnt(0);
        __asm__ volatile("" ::: "memory");   // TDM wrote LDS behind compiler's back

        // 3 rows x 256 f32, conflict-free ds_load_b128 reads.
        const f4* L = (const f4*)&lds[cur][wIdx][0];
        f4 a0 = L[lane +   0], a1 = L[lane +  32];
        f4 b0 = L[lane +  64], b1 = L[lane +  96];
        f4 c0 = L[lane + 128], c1 = L[lane + 160];

        f4 o0 = fma4(c0, w2, fma4(b0, w1, mul4(a0, w0)));
        f4 o1 = fma4(c1, w2, fma4(b1, w1, mul4(a1, w0)));

        // Non-temporal b128 stores: 64 MB write-once stream, keep L2 for tables.
        f4* o = (f4*)(out + (size_t)t * EMBED_DIM);
        __builtin_nontemporal_store(o0, o + lane);
        __builtin_nontemporal_store(o1, o + lane + 32);

        if (!more) break;
        t = tn; i0 = n0; i1 = n1; i2 = n2;
        cur ^= 1;
    }
}

extern "C" void kernel_launch(void* const* d_in, const int* in_sizes, int n_in,
                              void* d_out, int out_size, void* d_ws, size_t ws_size,
                              hipStream_t stream) {
    const int*   x   = (const int*)d_in[0];    // [B,S,H] int32
    const float* emb = (const float*)d_in[1];  // [10001, 256] f32
    const float* wt  = (const float*)d_in[2];  // [300003, 1] f32
    float*       out = (float*)d_out;          // [B,S,256] f32

    const int ntokens = in_sizes[0] / NUM_HASHES;   // 65536

    const int threads       = 32 * WAVES_PER_BLK;   // 256 = 8 wave32
    const int tokensPerWave = 8;                    // pipeline depth via grid-stride
    const int waves  = (ntokens + tokensPerWave - 1) / tokensPerWave;
    const int blocks = (waves + WAVES_PER_BLK - 1) / WAVES_PER_BLK;

    hipLaunchKernelGGL(hash_emb_bag_kernel, dim3(blocks), dim3(threads), 0, stream,
                       x, emb, wt, out, ntokens);
}